// SelectedUnitsHead_87806311399846
// MI455X (gfx1250) — compile-verified
//
#include <hip/hip_runtime.h>
#include <math.h>

// ---------------------------------------------------------------------------
// SelectedUnitsHead, MI455X (gfx1250, wave32).
//   K1 (fused, heterogeneous grid):
//     block 0   : func_embed / q / q2 / LSTM -> h[32]   (split-K VALU dots)
//     blocks 1-2: keym = relu(E[512,256] @ w_key[256,32] + b_key)
//                 via V_WMMA_F32_16X16X4_F32, w_key staged transposed in LDS
//   K2: scores, argmax, m = keym[units]/512, ar_out     (single WG)
// ---------------------------------------------------------------------------

typedef __attribute__((ext_vector_type(2))) float v2f;
typedef __attribute__((ext_vector_type(8))) float v8f;

#define WKT_STRIDE 258   // padded LDS row stride (dwords): even (b64 align), breaks bank conflicts

__global__ __launch_bounds__(512) void fused_stage1_kernel(
    const float* __restrict__ E,        // [512,256]
    const float* __restrict__ Wk,       // [256,32]
    const float* __restrict__ bk,       // [32]
    float* __restrict__ keym,           // [512,32]  (ws)
    const float* __restrict__ ar,       // [1024]
    const int*   __restrict__ binv,     // [512]
    const float* __restrict__ w_func,   // [512,256]
    const float* __restrict__ b_func,   // [256]
    const float* __restrict__ w_q1,     // [1024,256]
    const float* __restrict__ b_q1,     // [256]
    const float* __restrict__ w_q2,     // [256,32]
    const float* __restrict__ b_q2,     // [32]
    const float* __restrict__ w_lstm_k, // [32,128]
    const float* __restrict__ b_lstm,   // [128]
    float* __restrict__ h_out)          // [32]      (ws)
{
    __shared__ float smem[32 * WKT_STRIDE];   // 33 KB; keym: WkT, chain: scratch
    const int t = threadIdx.x;

    if (blockIdx.x == 0) {
        // ------------------- scalar head chain -> h[32] -------------------
        // 512 threads: output j = t&255, K-slice = t>>8 (2-way split-K).
        const int j = t & 255;
        const int slice = t >> 8;

        float fe = 0.f;   // partial of bin @ w_func
        for (int i = slice * 256; i < slice * 256 + 256; ++i)
            fe += (float)binv[i] * w_func[i * 256 + j];
        float qq = 0.f;   // partial of ar @ w_q1
        for (int i = slice * 512; i < slice * 512 + 512; ++i)
            qq += ar[i] * w_q1[i * 256 + j];

        smem[t] = fe;          // fe partials  [0..511]
        smem[512 + t] = qq;    // q partials   [512..1023]
        __syncthreads();

        if (t < 256) {
            float fev = smem[t] + smem[256 + t] + b_func[t];
            fev = fev > 0.f ? fev : 0.f;
            float qv = smem[512 + t] + smem[768 + t] + b_q1[t];
            qv = qv > 0.f ? qv : 0.f;
            smem[1024 + t] = fev + qv;        // s = func_embed + q
        }
        __syncthreads();

        if (t < 32) {                          // q2 = relu(s @ w_q2 + b_q2)
            float a = 0.f;
            for (int k = 0; k < 256; ++k)
                a += smem[1024 + k] * w_q2[k * 32 + t];
            a += b_q2[t];
            smem[1280 + t] = a > 0.f ? a : 0.f;
        }
        __syncthreads();

        if (t < 32) {
            // One LSTM step, h0=c0=0: recurrent kernel vanishes, f-gate unused.
            float zi = b_lstm[t];
            float zg = b_lstm[64 + t];
            float zo = b_lstm[96 + t];
            for (int k = 0; k < 32; ++k) {
                const float qk = smem[1280 + k];
                zi += qk * w_lstm_k[k * 128 + t];
                zg += qk * w_lstm_k[k * 128 + 64 + t];
                zo += qk * w_lstm_k[k * 128 + 96 + t];
            }
            const float si = 1.f / (1.f + __expf(-zi));
            const float so = 1.f / (1.f + __expf(-zo));
            const float c  = si * (zg > 0.f ? zg : 0.f);
            const float h  = so * (c > 0.f ? c : 0.f);
            h_out[t] = h;
        }
        return;
    }

    // ----------------------- keym via WMMA (blocks 1..2) ------------------
    // Stage w_key transposed into LDS: WkT[c][k] = Wk[k*32+c], stride 258.
    // Global reads coalesced; B fragments then become contiguous ds_load_b64.
    for (int idx = t; idx < 32 * 256; idx += 512) {
        const int k = idx >> 5;
        const int c = idx & 31;
        smem[c * WKT_STRIDE + k] = Wk[idx];
    }
    __syncthreads();

    const int w    = t >> 5;        // wave in block: 0..15
    const int lane = t & 31;
    const int half = lane >> 4;     // 0: K pair {0,1}, 1: K pair {2,3}
    const int lh   = lane & 15;
    const int tile = (blockIdx.x - 1) * 16 + w;   // M tile 0..31

    // A frag (16x4 f32): lane row = tile*16+lh; VGPR0/1 hold K {0,1}|{2,3} by lane half.
    const float* Erow = E + (tile * 16 + lh) * 256 + 2 * half;
    // B frags: contiguous K pairs from transposed LDS image.
    const float* B0 = smem + lh * WKT_STRIDE + 2 * half;          // N tile 0
    const float* B1 = smem + (16 + lh) * WKT_STRIDE + 2 * half;   // N tile 1

    v8f acc0 = {};
    v8f acc1 = {};
    for (int k0 = 0; k0 < 256; k0 += 4) {
        const float2 av  = *(const float2*)(Erow + k0);   // global_load_b64
        const float2 b0v = *(const float2*)(B0 + k0);     // ds_load_b64
        const float2 b1v = *(const float2*)(B1 + k0);     // ds_load_b64
        v2f a  = { av.x,  av.y  };
        v2f b0 = { b0v.x, b0v.y };
        v2f b1 = { b1v.x, b1v.y };
        acc0 = __builtin_amdgcn_wmma_f32_16x16x4_f32(
            false, a, false, b0, (short)0, acc0, false, false);
        acc1 = __builtin_amdgcn_wmma_f32_16x16x4_f32(
            false, a, false, b1, (short)0, acc1, false, false);
    }

    // D layout: VGPR v, lane l -> M = v + 8*(l>=16), N = l%16.
    const float bias0 = bk[lh];
    const float bias1 = bk[16 + lh];
#pragma unroll
    for (int v = 0; v < 8; ++v) {
        const int m = tile * 16 + v + 8 * half;
        float x0 = acc0[v] + bias0;
        float x1 = acc1[v] + bias1;
        keym[m * 32 + lh]      = x0 > 0.f ? x0 : 0.f;
        keym[m * 32 + 16 + lh] = x1 > 0.f ? x1 : 0.f;
    }
}

// ------------- Kernel 2: scores, argmax, selected unit, ar_out -------------
__global__ __launch_bounds__(512) void score_argmax_out_kernel(
    const float* __restrict__ keym,   // [512,32]
    const float* __restrict__ h,      // [32]
    const float* __restrict__ ar,     // [1024]
    const float* __restrict__ w_out,  // [32,1024]
    const float* __restrict__ b_out,  // [1024]
    float* __restrict__ out)          // [1537] = logits(512) | units(1) | ar_out(1024)
{
    __shared__ float s_h[32];
    __shared__ float s_val[512];
    __shared__ int   s_idx[512];
    __shared__ float s_m[32];
    const int t = threadIdx.x;

    if (t < 32) s_h[t] = h[t];
    __syncthreads();

    float sc = 0.f;
    const float* kr = keym + t * 32;
#pragma unroll
    for (int k = 0; k < 32; ++k)
        sc += kr[k] * s_h[k];
    out[t] = sc;              // units_logits
    s_val[t] = sc;
    s_idx[t] = t;
    __syncthreads();

    // argmax, first occurrence on ties (matches jnp.argmax)
    for (int off = 256; off > 0; off >>= 1) {
        if (t < off) {
            const float v1 = s_val[t + off];
            const int   i1 = s_idx[t + off];
            if (v1 > s_val[t] || (v1 == s_val[t] && i1 < s_idx[t])) {
                s_val[t] = v1;
                s_idx[t] = i1;
            }
        }
        __syncthreads();
    }
    const int units = s_idx[0];
    if (t == 0) out[512] = (float)units;

    // m = mean(onehot[:,None]*keym, axis=0) = keym[units,:] / 512
    if (t < 32) s_m[t] = keym[units * 32 + t] * (1.0f / 512.0f);
    __syncthreads();

#pragma unroll
    for (int rep = 0; rep < 2; ++rep) {
        const int o = t + rep * 512;
        float a = b_out[o];
#pragma unroll
        for (int k = 0; k < 32; ++k)
            a += s_m[k] * w_out[k * 1024 + o];
        a = a > 0.f ? a : 0.f;
        out[513 + o] = ar[o] + a;
    }
}

// ---------------------------------------------------------------------------
extern "C" void kernel_launch(void* const* d_in, const int* in_sizes, int n_in,
                              void* d_out, int out_size, void* d_ws, size_t ws_size,
                              hipStream_t stream) {
    (void)in_sizes; (void)n_in; (void)out_size; (void)ws_size;

    const float* ar       = (const float*)d_in[0];   // [1,1024]
    const int*   binv     = (const int*)  d_in[1];   // [1,512]
    const float* E        = (const float*)d_in[2];   // [1,512,256]
    const float* w_func   = (const float*)d_in[3];   // [512,256]
    const float* b_func   = (const float*)d_in[4];   // [256]
    const float* w_key    = (const float*)d_in[5];   // [256,32]
    const float* b_key    = (const float*)d_in[6];   // [32]
    const float* w_q1     = (const float*)d_in[7];   // [1024,256]
    const float* b_q1     = (const float*)d_in[8];   // [256]
    const float* w_q2     = (const float*)d_in[9];   // [256,32]
    const float* b_q2     = (const float*)d_in[10];  // [32]
    const float* w_lstm_k = (const float*)d_in[11];  // [32,128]
    /* d_in[12] w_lstm_r unused: h0 = 0 */
    const float* b_lstm   = (const float*)d_in[13];  // [128]
    const float* w_out    = (const float*)d_in[14];  // [32,1024]
    const float* b_out    = (const float*)d_in[15];  // [1024]

    float* out  = (float*)d_out;
    float* keym = (float*)d_ws;                         // 512*32*4 = 64 KB
    float* h    = (float*)((char*)d_ws + 64 * 1024);    // 32*4

    fused_stage1_kernel<<<3, 512, 0, stream>>>(E, w_key, b_key, keym,
                                               ar, binv, w_func, b_func,
                                               w_q1, b_q1, w_q2, b_q2,
                                               w_lstm_k, b_lstm, h);
    score_argmax_out_kernel<<<1, 512, 0, stream>>>(keym, h, ar, w_out, b_out, out);
}